// CorrelationLoss_79070347919431
// MI455X (gfx1250) — compile-verified
//
#include <hip/hip_runtime.h>
#include <math.h>

// ---------------------------------------------------------------------------
// CorrelationLoss for MI455X (gfx1250, wave32).
//
//   p_corr[n,s,l] = h[n,s] - (exp(p) @ p^T)[n,s,l],  h[n,s] = sum_c e^{p}*p
//   f_corr[n,s,l] = (fn @ fn^T)[n,s,l] - 0.3,        fn = f/max(||f||,1e-12)
//   pos = sum(relu(f_corr)*p_corr)/nnz(f_corr>0); neg analogous for f_corr<0
//
// Both GEMMs run on V_WMMA_F32_16X16X4_F32 (exact fp32, matches reference).
// Each wave register-blocks a 32x32 output region (2x2 WMMA tiles) so each
// K-step issues 4 WMMAs against 4 b64 loads (1:1) instead of 1:2.
// ---------------------------------------------------------------------------

typedef __attribute__((ext_vector_type(2))) float v2f;
typedef __attribute__((ext_vector_type(8))) float v8f;

#define NB   4      // batches
#define SPTS 1024   // 32*32 spatial points
#define CF   384    // feature channels
#define CK   32     // classes
#define BT   32     // per-wave output block (2x2 WMMA tiles)
#define NBT  (SPTS / BT)                // 32 blocks per axis
#define NBLK (NB * NBT * NBT)           // 4096 wave-blocks
#define DINO_SHIFT 0.3f

// ---------------------------------------------------------------- init ------
__global__ void cl_init_kernel(float* acc, unsigned int* cnt) {
    acc[0] = 0.f; acc[1] = 0.f;
    cnt[0] = 0u;  cnt[1] = 0u;
}

// ---------------------------------------------------------------- prep ------
// One wave (32 lanes) per spatial point: L2-normalize 384 feats, compute
// exp(p) and h = sum(exp(p)*p) over the 32 classes (one class per lane).
__global__ __launch_bounds__(256) void cl_prep_kernel(
    const float* __restrict__ feats, const float* __restrict__ p,
    float* __restrict__ fn, float* __restrict__ ep, float* __restrict__ h) {
    const int wave = (blockIdx.x * blockDim.x + threadIdx.x) >> 5;
    const int lane = threadIdx.x & 31;
    if (wave >= NB * SPTS) return;

    const float* f  = feats + (size_t)wave * CF;
    float*       fo = fn    + (size_t)wave * CF;

    float v[CF / 32];
    float ss = 0.f;
#pragma unroll
    for (int i = 0; i < CF / 32; ++i) {
        v[i] = f[lane + 32 * i];
        ss += v[i] * v[i];
    }
#pragma unroll
    for (int off = 16; off > 0; off >>= 1) ss += __shfl_down(ss, off, 32);
    ss = __shfl(ss, 0, 32);
    const float inv = 1.f / fmaxf(sqrtf(ss), 1e-12f);
#pragma unroll
    for (int i = 0; i < CF / 32; ++i) fo[lane + 32 * i] = v[i] * inv;

    const float pv = p[(size_t)wave * CK + lane];
    const float e  = expf(pv);
    ep[(size_t)wave * CK + lane] = e;
    float hp = e * pv;
#pragma unroll
    for (int off = 16; off > 0; off >>= 1) hp += __shfl_down(hp, off, 32);
    if (lane == 0) h[wave] = hp;
}

// ---------------------------------------------------------------- corr ------
// One wave per 32x32 output block (2x2 tiles of 16x16).
// f32 16x16x4 WMMA operand layout (wave32):
//   A (16x4): lanes 0-15 hold row M=lane, K={0,1}; lanes 16-31 row M=lane-16,
//             K={2,3}  -> per-lane float2 at row*ld + k + (lane>>4)*2.
//   B (4x16): transposed tile of the same row-major matrix -> identical
//             per-lane addressing with the column-tile base row.
//   C/D (16x16 f32, 8 VGPRs): VGPR r = row M=r (lanes 0-15) / M=r+8 (16-31).
__global__ __launch_bounds__(256) void cl_corr_kernel(
    const float* __restrict__ fn, const float* __restrict__ ep,
    const float* __restrict__ p,  const float* __restrict__ h,
    float* __restrict__ acc, unsigned int* __restrict__ cnt) {
    const int lane = threadIdx.x & 31;
    const int blk  = blockIdx.x * (blockDim.x >> 5) + (threadIdx.x >> 5);

    float pos = 0.f, neg = 0.f;
    unsigned int pcnt = 0u, ncnt = 0u;

    if (blk < NBLK) {                       // wave-uniform: EXEC stays all-1s
        const int n  = blk / (NBT * NBT);
        const int r  = blk % (NBT * NBT);
        const int s0 = (r / NBT) * BT;      // row block (s)
        const int l0 = (r % NBT) * BT;      // col block (l)

        const int mrow  = lane & 15;
        const int khalf = (lane >> 4) * 2;  // lanes 16-31 carry K+2..K+3

        // ---- Gram matrix: G = fn_s @ fn_l^T, K = 384, 2x2 tiles ------------
        const float* fA0 = fn + ((size_t)(n * SPTS + s0 + mrow)) * CF + khalf;
        const float* fA1 = fA0 + (size_t)16 * CF;
        const float* fB0 = fn + ((size_t)(n * SPTS + l0 + mrow)) * CF + khalf;
        const float* fB1 = fB0 + (size_t)16 * CF;
        v8f G00 = {}, G01 = {}, G10 = {}, G11 = {};
        for (int k = 0; k < CF; k += 4) {
            v2f a0 = *(const v2f*)(fA0 + k);
            v2f a1 = *(const v2f*)(fA1 + k);
            v2f b0 = *(const v2f*)(fB0 + k);
            v2f b1 = *(const v2f*)(fB1 + k);
            G00 = __builtin_amdgcn_wmma_f32_16x16x4_f32(false, a0, false, b0, (short)0, G00, false, false);
            G01 = __builtin_amdgcn_wmma_f32_16x16x4_f32(false, a0, false, b1, (short)0, G01, false, false);
            G10 = __builtin_amdgcn_wmma_f32_16x16x4_f32(false, a1, false, b0, (short)0, G10, false, false);
            G11 = __builtin_amdgcn_wmma_f32_16x16x4_f32(false, a1, false, b1, (short)0, G11, false, false);
        }

        // ---- JS term: Q = ep_s @ p_l^T, K = 32, 2x2 tiles ------------------
        const float* eA0 = ep + ((size_t)(n * SPTS + s0 + mrow)) * CK + khalf;
        const float* eA1 = eA0 + (size_t)16 * CK;
        const float* pB0 = p  + ((size_t)(n * SPTS + l0 + mrow)) * CK + khalf;
        const float* pB1 = pB0 + (size_t)16 * CK;
        v8f Q00 = {}, Q01 = {}, Q10 = {}, Q11 = {};
#pragma unroll
        for (int k = 0; k < CK; k += 4) {
            v2f a0 = *(const v2f*)(eA0 + k);
            v2f a1 = *(const v2f*)(eA1 + k);
            v2f b0 = *(const v2f*)(pB0 + k);
            v2f b1 = *(const v2f*)(pB1 + k);
            Q00 = __builtin_amdgcn_wmma_f32_16x16x4_f32(false, a0, false, b0, (short)0, Q00, false, false);
            Q01 = __builtin_amdgcn_wmma_f32_16x16x4_f32(false, a0, false, b1, (short)0, Q01, false, false);
            Q10 = __builtin_amdgcn_wmma_f32_16x16x4_f32(false, a1, false, b0, (short)0, Q10, false, false);
            Q11 = __builtin_amdgcn_wmma_f32_16x16x4_f32(false, a1, false, b1, (short)0, Q11, false, false);
        }

        // ---- fuse shift / relu masks / partial reduction -------------------
        // Tile row i covers s-rows [s0+16i, s0+16i+16); h is per-row (M only).
        const float* h0 = h + (size_t)n * SPTS + s0 + (lane >> 4) * 8;
        const float* h1 = h0 + 16;
#pragma unroll
        for (int rr = 0; rr < 8; ++rr) {
            const float hs0 = h0[rr];
            const float hs1 = h1[rr];

            float fc, pc;
            fc = G00[rr] - DINO_SHIFT; pc = hs0 - Q00[rr];
            if (fc > 0.f) { pos += fc * pc; ++pcnt; } else if (fc < 0.f) { neg += fc * pc; ++ncnt; }
            fc = G01[rr] - DINO_SHIFT; pc = hs0 - Q01[rr];
            if (fc > 0.f) { pos += fc * pc; ++pcnt; } else if (fc < 0.f) { neg += fc * pc; ++ncnt; }
            fc = G10[rr] - DINO_SHIFT; pc = hs1 - Q10[rr];
            if (fc > 0.f) { pos += fc * pc; ++pcnt; } else if (fc < 0.f) { neg += fc * pc; ++ncnt; }
            fc = G11[rr] - DINO_SHIFT; pc = hs1 - Q11[rr];
            if (fc > 0.f) { pos += fc * pc; ++pcnt; } else if (fc < 0.f) { neg += fc * pc; ++ncnt; }
        }
    }

#pragma unroll
    for (int off = 16; off > 0; off >>= 1) {
        pos  += __shfl_down(pos,  off, 32);
        neg  += __shfl_down(neg,  off, 32);
        pcnt += __shfl_down(pcnt, off, 32);
        ncnt += __shfl_down(ncnt, off, 32);
    }
    if (lane == 0) {
        atomicAdd(&acc[0], pos);
        atomicAdd(&acc[1], neg);
        atomicAdd(&cnt[0], pcnt);
        atomicAdd(&cnt[1], ncnt);
    }
}

// ------------------------------------------------------------- finalize -----
__global__ void cl_fin_kernel(const float* acc, const unsigned int* cnt,
                              float* out) {
    out[0] = acc[0] / (float)cnt[0];
    out[1] = acc[1] / (float)cnt[1];
}

// ---------------------------------------------------------------------------
extern "C" void kernel_launch(void* const* d_in, const int* in_sizes, int n_in,
                              void* d_out, int out_size, void* d_ws, size_t ws_size,
                              hipStream_t stream) {
    const float* feats = (const float*)d_in[0];   // [4,32,32,384] f32
    const float* p     = (const float*)d_in[1];   // [4,32,32,32]  f32 (log-probs)
    float* out = (float*)d_out;                   // [pos, neg]

    // workspace carve-up (all 8-byte aligned; total ~6.9 MB)
    float* fn  = (float*)d_ws;                    // 4*1024*384
    float* ep  = fn + (size_t)NB * SPTS * CF;     // 4*1024*32
    float* h   = ep + (size_t)NB * SPTS * CK;     // 4*1024
    float* acc = h + (size_t)NB * SPTS;           // 2 floats
    unsigned int* cnt = (unsigned int*)(acc + 2); // 2 uints

    cl_init_kernel<<<1, 1, 0, stream>>>(acc, cnt);

    // one wave per point: 4096 waves / 8 waves-per-block
    cl_prep_kernel<<<(NB * SPTS) / 8, 256, 0, stream>>>(feats, p, fn, ep, h);

    // one wave per 32x32 output block: 4096 waves / 8 waves-per-block
    cl_corr_kernel<<<NBLK / 8, 256, 0, stream>>>(fn, ep, p, h, acc, cnt);

    cl_fin_kernel<<<1, 1, 0, stream>>>(acc, cnt, out);
}